// Zernike3Deep_43044162240770
// MI455X (gfx1250) — compile-verified
//
#include <hip/hip_runtime.h>

typedef __attribute__((ext_vector_type(2))) float v2f;
typedef __attribute__((ext_vector_type(8))) float v8f;

// Problem constants (fixed by the reference)
constexpr int Bn   = 8;
constexpr int LATn = 128;
constexpr int Hn   = 1024;
constexpr int Kn   = 294;
constexpr int Nn   = 100000;
constexpr int KPAD = 304;   // K padded to multiple of 4 (and 16)
constexpr int AST  = 305;   // LDS stride for A tile (odd -> conflict-free)
constexpr int KMAIN = 292;  // largest k0 bound with all rows < Kn (k0+3 <= 293)

// ---------------------------------------------------------------------------
// Kernel 1: full MLP decode (single workgroup).  Produces:
//   Acoef[32][304]  (rows 0-7 = 64*cx, 8-15 = 64*cy, 16-23 = 64*cz, rest 0)
//   reg scalar -> reg_out[0]
// ---------------------------------------------------------------------------
__global__ __launch_bounds__(512)
void mlp_kernel(const float* __restrict__ latent,
                const float* __restrict__ W0, const float* __restrict__ b0,
                const float* __restrict__ W1, const float* __restrict__ b1,
                const float* __restrict__ W2, const float* __restrict__ b2,
                const float* __restrict__ W3, const float* __restrict__ b3,
                const float* __restrict__ Wx, const float* __restrict__ bx,
                const float* __restrict__ Wy, const float* __restrict__ by,
                const float* __restrict__ Wz, const float* __restrict__ bz,
                float* __restrict__ Acoef, float* __restrict__ reg_out)
{
    __shared__ float lat[Bn * LATn];    // 4 KB
    __shared__ float h[Bn * Hn];        // 32 KB
    __shared__ float red[512];

    const int tid = threadIdx.x;

    for (int i = tid; i < Bn * LATn; i += 512) lat[i] = latent[i];
    __syncthreads();

    // layer 0: h = relu(latent @ W0 + b0)
    for (int j = tid; j < Hn; j += 512) {
        float acc[Bn];
        #pragma unroll
        for (int b = 0; b < Bn; ++b) acc[b] = 0.f;
        for (int k = 0; k < LATn; ++k) {
            const float w = W0[k * Hn + j];
            #pragma unroll
            for (int b = 0; b < Bn; ++b) acc[b] += lat[b * LATn + k] * w;
        }
        const float bb = b0[j];
        #pragma unroll
        for (int b = 0; b < Bn; ++b) h[b * Hn + j] = fmaxf(acc[b] + bb, 0.f);
    }
    __syncthreads();

    // 3 residual layers: h = relu(h + h @ W + b)
    const float* Ws[3] = {W1, W2, W3};
    const float* bs[3] = {b1, b2, b3};
    for (int L = 0; L < 3; ++L) {
        const float* W = Ws[L];
        const float* bv = bs[L];
        float nacc[2][Bn];
        float oldv[2][Bn];
        float bias[2];
        #pragma unroll
        for (int c = 0; c < 2; ++c) {
            const int j = tid + c * 512;
            #pragma unroll
            for (int b = 0; b < Bn; ++b) nacc[c][b] = 0.f;
            for (int k = 0; k < Hn; ++k) {
                const float w = W[k * Hn + j];
                #pragma unroll
                for (int b = 0; b < Bn; ++b) nacc[c][b] += h[b * Hn + k] * w;
            }
            #pragma unroll
            for (int b = 0; b < Bn; ++b) oldv[c][b] = h[b * Hn + j];
            bias[c] = bv[j];
        }
        __syncthreads();    // all reads of old h done
        #pragma unroll
        for (int c = 0; c < 2; ++c) {
            const int j = tid + c * 512;
            #pragma unroll
            for (int b = 0; b < Bn; ++b)
                h[b * Hn + j] = fmaxf(oldv[c][b] + nacc[c][b] + bias[c], 0.f);
        }
        __syncthreads();    // writes visible before next layer
    }

    // Heads: cx/cy/cz = h @ W{x,y,z} + b{x,y,z}; build padded A matrix
    float sumsq = 0.f;
    for (int j = tid; j < KPAD; j += 512) {
        if (j < Kn) {
            float ax[Bn], ay[Bn], az[Bn];
            #pragma unroll
            for (int b = 0; b < Bn; ++b) { ax[b] = 0.f; ay[b] = 0.f; az[b] = 0.f; }
            for (int k = 0; k < Hn; ++k) {
                const float wx = Wx[k * Kn + j];
                const float wy = Wy[k * Kn + j];
                const float wz = Wz[k * Kn + j];
                #pragma unroll
                for (int b = 0; b < Bn; ++b) {
                    const float hv = h[b * Hn + k];
                    ax[b] += hv * wx; ay[b] += hv * wy; az[b] += hv * wz;
                }
            }
            const float bxv = bx[j], byv = by[j], bzv = bz[j];
            #pragma unroll
            for (int b = 0; b < Bn; ++b) {
                const float cx = ax[b] + bxv;
                const float cy = ay[b] + byv;
                const float cz = az[b] + bzv;
                sumsq += cx * cx + cy * cy + cz * cz;
                // fold 0.5*XSIZE = 64 into the coefficient matrix
                Acoef[(b)      * KPAD + j] = 64.f * cx;
                Acoef[(8 + b)  * KPAD + j] = 64.f * cy;
                Acoef[(16 + b) * KPAD + j] = 64.f * cz;
            }
            #pragma unroll
            for (int r = 24; r < 32; ++r) Acoef[r * KPAD + j] = 0.f;
        } else {
            #pragma unroll
            for (int r = 0; r < 32; ++r) Acoef[r * KPAD + j] = 0.f;
        }
    }

    red[tid] = sumsq;
    __syncthreads();
    for (int s = 256; s > 0; s >>= 1) {
        if (tid < s) red[tid] += red[tid + s];
        __syncthreads();
    }
    if (tid == 0) reg_out[0] = 1e-4f * sqrtf(red[0]);
}

// ---------------------------------------------------------------------------
// Kernel 2: flow0[n][b][c] = (64 * c) @ Z  via V_WMMA_F32_16X16X4_F32.
// Each wave owns 32 columns (nodes): 2 row-tiles x 2 col-tiles = 4 wmma
// accumulators; every Z element loaded once, A fragments reused across the
// two column tiles.  No divergence before the WMMA loop ends (EXEC all-1s).
// ---------------------------------------------------------------------------
__global__ __launch_bounds__(256)
void einsum_kernel(const float* __restrict__ Acoef,
                   const float* __restrict__ Z,
                   float* __restrict__ f0)
{
    __shared__ float As[32 * AST];   // ~39 KB, conflict-free odd stride

    const int tid = threadIdx.x;
    for (int i = tid; i < 32 * KPAD; i += 256) {   // 9728/256 = 38 uniform iters
        const int r = i / KPAD;
        const int c = i - r * KPAD;
        As[r * AST + c] = Acoef[i];
    }
    __syncthreads();

    const int wave = tid >> 5;
    const int lane = tid & 31;
    const int mrow = lane & 15;
    const int koff = (lane < 16) ? 0 : 2;        // K sub-block per lane half

    const int n0  = blockIdx.x * 256 + wave * 32 + mrow;  // column tile 0
    const int n1  = n0 + 16;                              // column tile 1
    const int nc0 = (n0 < Nn) ? n0 : (Nn - 1);
    const int nc1 = (n1 < Nn) ? n1 : (Nn - 1);

    v8f acc00 = {};   // rows 0-15  x cols n0..n0+15   (cx | cy)
    v8f acc10 = {};   // rows 16-31 x cols n0..n0+15   (cz | pad)
    v8f acc01 = {};   // rows 0-15  x cols n1..n1+15
    v8f acc11 = {};   // rows 16-31 x cols n1..n1+15

    const float* a0p = As + mrow * AST + koff;
    const float* a1p = As + (mrow + 16) * AST + koff;

    // main loop: rows k0+koff, k0+koff+1 all < Kn for k0 < KMAIN
    const float* zp0 = Z + (size_t)koff * Nn + nc0;
    const float* zp1 = Z + (size_t)koff * Nn + nc1;
    for (int k0 = 0; k0 < KMAIN; k0 += 4) {
        v2f a0, a1, bv0, bv1;
        a0.x = a0p[k0];  a0.y = a0p[k0 + 1];
        a1.x = a1p[k0];  a1.y = a1p[k0 + 1];
        bv0.x = zp0[0];  bv0.y = zp0[Nn];
        bv1.x = zp1[0];  bv1.y = zp1[Nn];
        acc00 = __builtin_amdgcn_wmma_f32_16x16x4_f32(false, a0, false, bv0,
                                                      (short)0, acc00, false, false);
        acc10 = __builtin_amdgcn_wmma_f32_16x16x4_f32(false, a1, false, bv0,
                                                      (short)0, acc10, false, false);
        acc01 = __builtin_amdgcn_wmma_f32_16x16x4_f32(false, a0, false, bv1,
                                                      (short)0, acc01, false, false);
        acc11 = __builtin_amdgcn_wmma_f32_16x16x4_f32(false, a1, false, bv1,
                                                      (short)0, acc11, false, false);
        zp0 += 4 * (size_t)Nn;
        zp1 += 4 * (size_t)Nn;
    }

    // tail: k0 = 292, 296, 300 — clamp row index; A columns >= Kn are zero,
    // so clamped (finite) B values contribute exactly 0.  Selects, no branches.
    for (int k0 = KMAIN; k0 < KPAD; k0 += 4) {
        const int k   = k0 + koff;
        const int kz0 = (k     < Kn) ? k     : (Kn - 1);
        const int kz1 = (k + 1 < Kn) ? k + 1 : (Kn - 1);
        v2f a0, a1, bv0, bv1;
        a0.x = a0p[k0];  a0.y = a0p[k0 + 1];
        a1.x = a1p[k0];  a1.y = a1p[k0 + 1];
        bv0.x = Z[(size_t)kz0 * Nn + nc0];  bv0.y = Z[(size_t)kz1 * Nn + nc0];
        bv1.x = Z[(size_t)kz0 * Nn + nc1];  bv1.y = Z[(size_t)kz1 * Nn + nc1];
        acc00 = __builtin_amdgcn_wmma_f32_16x16x4_f32(false, a0, false, bv0,
                                                      (short)0, acc00, false, false);
        acc10 = __builtin_amdgcn_wmma_f32_16x16x4_f32(false, a1, false, bv0,
                                                      (short)0, acc10, false, false);
        acc01 = __builtin_amdgcn_wmma_f32_16x16x4_f32(false, a0, false, bv1,
                                                      (short)0, acc01, false, false);
        acc11 = __builtin_amdgcn_wmma_f32_16x16x4_f32(false, a1, false, bv1,
                                                      (short)0, acc11, false, false);
    }

    // store: D tile lane L holds N=lane%16, VGPR v -> M = v + (lane<16 ? 0 : 8)
    const bool lo = (lane < 16);
    if (n0 < Nn) {
        float* base = f0 + (size_t)n0 * 24;   // [n][b][c] node-major
        #pragma unroll
        for (int v = 0; v < 8; ++v) {
            if (lo) base[v * 3 + 0] = acc00[v];   // M=v    -> cx, b=v
            else    base[v * 3 + 1] = acc00[v];   // M=8+v  -> cy, b=v
        }
        if (lo) {
            #pragma unroll
            for (int v = 0; v < 8; ++v)
                base[v * 3 + 2] = acc10[v];       // M=16+v -> cz, b=v
        }
    }
    if (n1 < Nn) {
        float* base = f0 + (size_t)n1 * 24;
        #pragma unroll
        for (int v = 0; v < 8; ++v) {
            if (lo) base[v * 3 + 0] = acc01[v];
            else    base[v * 3 + 1] = acc01[v];
        }
        if (lo) {
            #pragma unroll
            for (int v = 0; v < 8; ++v)
                base[v * 3 + 2] = acc11[v];
        }
    }
}

// ---------------------------------------------------------------------------
// Diffusion kernels
// ---------------------------------------------------------------------------
__global__ void fill_zero_kernel(float* __restrict__ p, int n)
{
    const int i = blockIdx.x * blockDim.x + threadIdx.x;
    if (i < n) p[i] = 0.f;
}

__global__ void deg_kernel(const float* __restrict__ ew,
                           const long long* __restrict__ dst,
                           float* __restrict__ deg, int E)
{
    const int e = blockIdx.x * blockDim.x + threadIdx.x;
    if (e >= E) return;
    atomicAdd(deg + dst[e], ew[e]);
}

__global__ void scatter_kernel(const float* __restrict__ f,
                               const float* __restrict__ ew,
                               const long long* __restrict__ src,
                               const long long* __restrict__ dst,
                               float* __restrict__ agg, int E)
{
    const int e = blockIdx.x * blockDim.x + threadIdx.x;
    if (e >= E) return;
    const long long s = src[e];
    const long long d = dst[e];
    const float w = ew[e];
    const float4* fv = reinterpret_cast<const float4*>(f + (size_t)s * 24);
    float* ad = agg + (size_t)d * 24;
    #pragma unroll
    for (int q = 0; q < 6; ++q) {
        const float4 v = fv[q];
        atomicAdd(ad + q * 4 + 0, v.x * w);
        atomicAdd(ad + q * 4 + 1, v.y * w);
        atomicAdd(ad + q * 4 + 2, v.z * w);
        atomicAdd(ad + q * 4 + 3, v.w * w);
    }
}

__global__ void combine_kernel(const float* __restrict__ f,
                               const float* __restrict__ agg,
                               const float* __restrict__ deg,
                               float* __restrict__ fn)
{
    const int gid = blockIdx.x * blockDim.x + threadIdx.x;
    if (gid >= Nn * 6) return;
    const int nidx = gid / 6;
    const int q = gid - nidx * 6;
    const float s = 0.5f / fmaxf(deg[nidx], 1e-6f);
    const float4 fv = reinterpret_cast<const float4*>(f   + (size_t)nidx * 24)[q];
    const float4 av = reinterpret_cast<const float4*>(agg + (size_t)nidx * 24)[q];
    float4 o;
    o.x = 0.5f * fv.x + s * av.x;
    o.y = 0.5f * fv.y + s * av.y;
    o.z = 0.5f * fv.z + s * av.z;
    o.w = 0.5f * fv.w + s * av.w;
    reinterpret_cast<float4*>(fn + (size_t)nidx * 24)[q] = o;
}

// Final step: combine + transpose [N][B][3] -> [B][N][3] directly into d_out
__global__ void combine_final_kernel(const float* __restrict__ f,
                                     const float* __restrict__ agg,
                                     const float* __restrict__ deg,
                                     float* __restrict__ out)
{
    const int n = blockIdx.x * blockDim.x + threadIdx.x;
    if (n >= Nn) return;
    const float s = 0.5f / fmaxf(deg[n], 1e-6f);
    const float* fb = f   + (size_t)n * 24;
    const float* ab = agg + (size_t)n * 24;
    #pragma unroll
    for (int b = 0; b < 8; ++b) {
        #pragma unroll
        for (int c = 0; c < 3; ++c) {
            out[(size_t)b * (3 * (size_t)Nn) + (size_t)n * 3 + c] =
                0.5f * fb[b * 3 + c] + s * ab[b * 3 + c];
        }
    }
}

// ---------------------------------------------------------------------------
extern "C" void kernel_launch(void* const* d_in, const int* in_sizes, int n_in,
                              void* d_out, int out_size, void* d_ws, size_t ws_size,
                              hipStream_t stream)
{
    const float* latent = (const float*)d_in[0];
    const float* W0 = (const float*)d_in[1];
    const float* b0 = (const float*)d_in[2];
    const float* W1 = (const float*)d_in[3];
    const float* b1 = (const float*)d_in[4];
    const float* W2 = (const float*)d_in[5];
    const float* b2 = (const float*)d_in[6];
    const float* W3 = (const float*)d_in[7];
    const float* b3 = (const float*)d_in[8];
    const float* Wx = (const float*)d_in[9];
    const float* bx = (const float*)d_in[10];
    const float* Wy = (const float*)d_in[11];
    const float* by = (const float*)d_in[12];
    const float* Wz = (const float*)d_in[13];
    const float* bz = (const float*)d_in[14];
    const float* Z  = (const float*)d_in[15];
    const float* ew = (const float*)d_in[16];
    const long long* ei = (const long long*)d_in[17];
    const int E = in_sizes[16];

    float* out = (float*)d_out;

    // workspace layout (floats)
    float* wsf   = (float*)d_ws;
    float* Acoef = wsf;                       // 32*304 = 9728
    float* fA    = wsf + 10240;               // [N][24] = 2,400,000
    float* agg   = fA + (size_t)Nn * 24;      // [N][24]
    float* deg   = agg + (size_t)Nn * 24;     // [N]
    float* fB    = out;                       // reuse d_out flow region as scratch
    float* reg_out = out + (size_t)Bn * Nn * 3;

    const long long* src = ei;
    const long long* dst = ei + E;

    const int TB = 256;
    const int egrid = (E + TB - 1) / TB;

    // 1. MLP decode -> Acoef (+ reg)
    mlp_kernel<<<1, 512, 0, stream>>>(latent, W0, b0, W1, b1, W2, b2, W3, b3,
                                      Wx, bx, Wy, by, Wz, bz, Acoef, reg_out);

    // 2. WMMA einsum -> fA ([N][24], scale folded in)
    einsum_kernel<<<(Nn + 255) / 256, 256, 0, stream>>>(Acoef, Z, fA);

    // 3. degree
    fill_zero_kernel<<<(Nn + TB - 1) / TB, TB, 0, stream>>>(deg, Nn);
    deg_kernel<<<egrid, TB, 0, stream>>>(ew, dst, deg, E);

    // 4. diffusion step 1: fA -> fB
    fill_zero_kernel<<<(Nn * 24 + TB - 1) / TB, TB, 0, stream>>>(agg, Nn * 24);
    scatter_kernel<<<egrid, TB, 0, stream>>>(fA, ew, src, dst, agg, E);
    combine_kernel<<<(Nn * 6 + TB - 1) / TB, TB, 0, stream>>>(fA, agg, deg, fB);

    // 5. diffusion step 2: fB -> fA
    fill_zero_kernel<<<(Nn * 24 + TB - 1) / TB, TB, 0, stream>>>(agg, Nn * 24);
    scatter_kernel<<<egrid, TB, 0, stream>>>(fB, ew, src, dst, agg, E);
    combine_kernel<<<(Nn * 6 + TB - 1) / TB, TB, 0, stream>>>(fB, agg, deg, fA);

    // 6. diffusion step 3: fA -> out ([B][N][3])
    fill_zero_kernel<<<(Nn * 24 + TB - 1) / TB, TB, 0, stream>>>(agg, Nn * 24);
    scatter_kernel<<<egrid, TB, 0, stream>>>(fA, ew, src, dst, agg, E);
    combine_final_kernel<<<(Nn + TB - 1) / TB, TB, 0, stream>>>(fA, agg, deg, out);
}